// MultichannelMultiheadAttention_3633542332462
// MI455X (gfx1250) — compile-verified
//
#include <hip/hip_runtime.h>

// ---------------------------------------------------------------------------
// MI455X (gfx1250): fused conv3x3(q,k,v) -> fp32-WMMA per-channel GEMMs
// (double-buffered GLOBAL_LOAD_ASYNC_TO_LDS pipeline) -> rotary -> WMMA QK^T
// (+prev) -> softmax + WMMA P*V -> WMMA output projection.
// All matrix math uses V_WMMA_F32_16X16X4_F32 (exact fp32, wave32).  Problem
// is HBM-bound (~470MB / 23.3TB/s ~ 20us); fp32 WMMA is far above the memory
// floor, so the work here is data movement: async B128 copies into LDS
// (ASYNCcnt), conflict-free padded layouts, coalesced 16B global runs.
// ---------------------------------------------------------------------------

typedef float v2f __attribute__((ext_vector_type(2)));
typedef float v8f __attribute__((ext_vector_type(8)));

#define Bn  2
#define Cn  8
#define Fn  1024
#define Wn  512
#define NHn 8
#define HDn 128

static __device__ inline v8f vzero8() {
  v8f z = {0.f,0.f,0.f,0.f,0.f,0.f,0.f,0.f};
  return z;
}

// fp32 WMMA: D = A(16x4) * B(4x16) + C
static __device__ inline v8f wmma_f32(v2f a, v2f b, v8f c) {
  return __builtin_amdgcn_wmma_f32_16x16x4_f32(false, a, false, b, (short)0, c,
                                               false, false);
}

// Low 32 bits of a generic pointer to __shared__ = LDS byte offset
// (shared aperture occupies addr[63:32] on gfx1250).
static __device__ inline unsigned lds_off(const void* p) {
  return (unsigned)(unsigned long long)p;
}

// CDNA5 async global->LDS copies (ASYNCcnt).  GV mode: 64-bit addr in VGPR
// pair, LDS byte offset in VGPR.
static __device__ inline void async_cp_b32(unsigned lds, const void* g) {
  asm volatile("global_load_async_to_lds_b32 %0, %1, off"
               :: "v"(lds), "v"(g) : "memory");
}
static __device__ inline void async_cp_b128(unsigned lds, const void* g) {
  asm volatile("global_load_async_to_lds_b128 %0, %1, off"
               :: "v"(lds), "v"(g) : "memory");
}
static __device__ inline void wait_async0() {
  asm volatile("s_wait_asynccnt 0x0" ::: "memory");
}

// ---------------------------------------------------------------------------
// Kernel 1: fused 3x3 conv for q,k,v.  co is block-uniform -> scalar weights;
// x loads shared across the three accumulators.
// ---------------------------------------------------------------------------
__global__ __launch_bounds__(256) void conv3x3_qkv(
    const float* __restrict__ x,
    const float* __restrict__ wq, const float* __restrict__ bq,
    const float* __restrict__ wk, const float* __restrict__ bk,
    const float* __restrict__ wv, const float* __restrict__ bv,
    float* __restrict__ qc, float* __restrict__ kc, float* __restrict__ vc)
{
  const int xw = blockIdx.x * 256 + threadIdx.x;   // 0..511
  const int y  = blockIdx.y;                       // 0..1023
  const int b  = blockIdx.z / Cn;
  const int co = blockIdx.z % Cn;
  float aq = bq[co], ak = bk[co], av = bv[co];
  for (int ci = 0; ci < Cn; ++ci) {
    const float* xp = x + ((size_t)(b * Cn + ci) * Fn) * Wn;
    const int wb = (co * Cn + ci) * 9;
    #pragma unroll
    for (int ky = 0; ky < 3; ++ky) {
      const int yy = y + ky - 1;
      if (yy < 0 || yy >= Fn) continue;
      #pragma unroll
      for (int kx = 0; kx < 3; ++kx) {
        const int xx = xw + kx - 1;
        if (xx < 0 || xx >= Wn) continue;
        const float xv = xp[(size_t)yy * Wn + xx];
        aq = fmaf(xv, wq[wb + ky * 3 + kx], aq);
        ak = fmaf(xv, wk[wb + ky * 3 + kx], ak);
        av = fmaf(xv, wv[wb + ky * 3 + kx], av);
      }
    }
  }
  const size_t o = ((size_t)(b * Cn + co) * Fn + y) * Wn + xw;
  qc[o] = aq; kc[o] = ak; vc[o] = av;
}

// ---------------------------------------------------------------------------
// Kernel 2: per-channel linear  D[f,w] = sum_h L[c,f,h] * S[bc,h,w]
// M=1024, N=512, K=1024 per (b,c).  Macro-tile 64(M) x 128(N), K-chunk 16,
// DOUBLE-BUFFERED async global->LDS B128 pipeline:
//   issue chunk k+1 async -> compute chunk k from LDS -> s_wait_asynccnt 0 ->
//   barrier -> swap.
// A_lds [64][20]: rows 80B (16B-aligned for B128); banks 4*(5L%16) distinct
//   in-half, cross-half 20*dL==2 (mod 64) impossible -> conflict-free.
// B_lds [16][136]: rows 544B; 2*136 mod 64 = 16 -> cross-half conflict-free.
// ---------------------------------------------------------------------------
#define APAD 20
#define BPAD 136
__global__ __launch_bounds__(256) void mc_linear_wmma(
    const float* __restrict__ lin,   // (C, F, F)
    const float* __restrict__ src,   // (B*C, F, W)
    float* __restrict__ dst)         // (B*C, F, W)
{
  __shared__ float Asub[2][64 * APAD];
  __shared__ float Bsub[2][16 * BPAD];
  const int tid  = threadIdx.x;
  const int wv   = tid >> 5;
  const int lane = tid & 31;
  const int L    = lane & 15;
  const int hf   = lane >> 4;           // lane half (0/1)
  const int mbase = blockIdx.x * 64;
  const int nbase = blockIdx.y * 128;
  const int bc    = blockIdx.z;
  const int c     = bc % Cn;
  const float* A  = lin + (size_t)c * Fn * Fn;   // [f][h], row stride F
  const float* Bp = src + (size_t)bc * Fn * Wn;  // [h][w]
  float*       D  = dst + (size_t)bc * Fn * Wn;

  // Per-thread async-copy coordinates (16B granules).
  const int am = tid >> 2;              // A row (0..63)
  const int ak = (tid & 3) * 4;         // A k-granule
  v8f acc[4];
  #pragma unroll
  for (int t = 0; t < 4; ++t) acc[t] = vzero8();

  // A: 64x16 = 256 granules -> 1 B128/thread.  B: 16x128 = 512 -> 2/thread.
  #define ISSUE_CHUNK(buf, kb)                                                 \
    do {                                                                       \
      async_cp_b128(lds_off(&Asub[buf][am * APAD + ak]),                       \
                    &A[(size_t)(mbase + am) * Fn + (kb) + ak]);                \
      _Pragma("unroll")                                                        \
      for (int i_ = 0; i_ < 2; ++i_) {                                         \
        const int e_ = tid + i_ * 256;                                         \
        const int bk_ = e_ >> 5, bn_ = (e_ & 31) * 4;                          \
        async_cp_b128(lds_off(&Bsub[buf][bk_ * BPAD + bn_]),                   \
                      &Bp[(size_t)((kb) + bk_) * Wn + nbase + bn_]);           \
      }                                                                        \
    } while (0)

  ISSUE_CHUNK(0, 0);
  wait_async0();
  __syncthreads();

  int cur = 0;
  for (int kb = 0; kb < Fn; kb += 16) {
    const int nxt = cur ^ 1;
    if (kb + 16 < Fn) ISSUE_CHUNK(nxt, kb + 16);   // overlap with compute
    const float* As = Asub[cur];
    const float* Bs = Bsub[cur];
    #pragma unroll
    for (int ks = 0; ks < 4; ++ks) {
      const int k0 = ks * 4 + hf * 2;
      v2f bf;
      bf.x = Bs[(k0 + 0) * BPAD + wv * 16 + L];
      bf.y = Bs[(k0 + 1) * BPAD + wv * 16 + L];
      #pragma unroll
      for (int mt = 0; mt < 4; ++mt) {
        v2f af;
        af.x = As[(mt * 16 + L) * APAD + k0 + 0];
        af.y = As[(mt * 16 + L) * APAD + k0 + 1];
        acc[mt] = wmma_f32(af, bf, acc[mt]);
      }
    }
    wait_async0();       // own copies done...
    __syncthreads();     // ...and everyone's, before reading nxt / reusing cur
    cur = nxt;
  }
  #undef ISSUE_CHUNK

  #pragma unroll
  for (int mt = 0; mt < 4; ++mt) {
    #pragma unroll
    for (int j = 0; j < 8; ++j) {          // C/D: M = j + 8*half, N = lane%16
      const int m = mbase + mt * 16 + hf * 8 + j;
      const int n = nbase + wv * 16 + L;
      D[(size_t)m * Wn + n] = acc[mt][j];
    }
  }
}

// ---------------------------------------------------------------------------
// Kernel 3: interleaved rotary, in-place on q_p and k_p (one launch).
// Pair (f0, f0+1) within a head rotates by angle w * 10000^(-(f0%128)/128).
// ---------------------------------------------------------------------------
__global__ __launch_bounds__(256) void rotary_inplace(
    float* __restrict__ qp, float* __restrict__ kp)
{
  const long long PAIRS = (long long)Bn * Cn * (Fn / 2) * Wn;   // 4,194,304
  long long idx = (long long)blockIdx.x * 256 + threadIdx.x;
  float* buf = qp;
  if (idx >= PAIRS) { buf = kp; idx -= PAIRS; }
  const int w  = (int)(idx % Wn);
  const long long t = idx / Wn;
  const int fp = (int)(t % (Fn / 2));
  const int bc = (int)(t / (Fn / 2));
  const int f0 = fp * 2;
  const float expo = -(float)(f0 & (HDn - 1)) / (float)HDn;
  const float ang  = (float)w * __powf(10000.0f, expo);
  float s, c;
  __sincosf(ang, &s, &c);
  const size_t o1 = ((size_t)bc * Fn + f0) * Wn + w;
  const size_t o2 = o1 + Wn;
  const float x1 = buf[o1], x2 = buf[o2];
  buf[o1] = x1 * c - x2 * s;
  buf[o2] = x2 * c + x1 * s;
}

// ---------------------------------------------------------------------------
// Kernel 4: qk = Q K^T / 32 + prev_qk per head.  M=N=512, K=128 (d).
// Macro 64x64, d-chunked by 64.  LDS tiles are d-MAJOR [64 d][72 m] so both
// global and LDS runs are m-contiguous -> async B128 staging (4/thread/tile).
// Stride 72: rows 288B (16B-aligned); frag reads addr = k0*72 + base + L ->
// in-half banks consecutive; cross-half 2*72 mod 64 = 16 -> conflict-free.
// ---------------------------------------------------------------------------
#define QMS 72
__global__ __launch_bounds__(256) void qk_wmma(
    const float* __restrict__ qp, const float* __restrict__ kp,
    const float* __restrict__ prev, float* __restrict__ qkout)
{
  __shared__ float Qs[64 * QMS];   // [d][m]
  __shared__ float Ks[64 * QMS];   // [d][m]
  const int tid  = threadIdx.x;
  const int wv   = tid >> 5, lane = tid & 31, L = lane & 15, hf = lane >> 4;
  const int mb   = blockIdx.x * 64;
  const int nb   = blockIdx.y * 64;
  const int head = blockIdx.z;                  // (b*C+c)*NH + n
  const int bc   = head >> 3, nh = head & 7;
  const float* qb = qp + (size_t)bc * Fn * Wn + (size_t)nh * HDn * Wn; // [d][pos]
  const float* kb = kp + (size_t)bc * Fn * Wn + (size_t)nh * HDn * Wn;
  const float* pb = prev  + (size_t)head * Wn * Wn;
  float*       ob = qkout + (size_t)head * Wn * Wn;

  const int mt  = wv & 3;
  const int ntb = (wv >> 2) * 2;
  v8f acc0 = vzero8(), acc1 = vzero8();

  for (int db = 0; db < HDn; db += 64) {
    __syncthreads();                       // WAR vs previous chunk's compute
    #pragma unroll
    for (int i = 0; i < 4; ++i) {          // 64d x 64m per tile, B128 granules
      const int e  = tid + i * 256;
      const int m4 = (e & 15) * 4, d = e >> 4;
      async_cp_b128(lds_off(&Qs[d * QMS + m4]),
                    &qb[(size_t)(db + d) * Wn + mb + m4]);
      async_cp_b128(lds_off(&Ks[d * QMS + m4]),
                    &kb[(size_t)(db + d) * Wn + nb + m4]);
    }
    wait_async0();
    __syncthreads();
    #pragma unroll 4
    for (int ks = 0; ks < 16; ++ks) {
      const int k0 = ks * 4 + hf * 2;
      v2f af, b0, b1;
      af.x = Qs[(k0 + 0) * QMS + mt * 16 + L];
      af.y = Qs[(k0 + 1) * QMS + mt * 16 + L];
      b0.x = Ks[(k0 + 0) * QMS + ntb * 16 + L];
      b0.y = Ks[(k0 + 1) * QMS + ntb * 16 + L];
      b1.x = Ks[(k0 + 0) * QMS + (ntb + 1) * 16 + L];
      b1.y = Ks[(k0 + 1) * QMS + (ntb + 1) * 16 + L];
      acc0 = wmma_f32(af, b0, acc0);
      acc1 = wmma_f32(af, b1, acc1);
    }
  }
  const float scale = 1.0f / 32.0f;        // 1/sqrt(F=1024), as in reference
  #pragma unroll
  for (int j = 0; j < 8; ++j) {
    const int q  = mb + mt * 16 + hf * 8 + j;
    const int c0 = nb + ntb * 16 + L;
    const int c1 = nb + (ntb + 1) * 16 + L;
    ob[(size_t)q * Wn + c0] = acc0[j] * scale + pb[(size_t)q * Wn + c0];
    ob[(size_t)q * Wn + c1] = acc1[j] * scale + pb[(size_t)q * Wn + c1];
  }
}

// ---------------------------------------------------------------------------
// Kernel 5: softmax(qk) @ V per head, writing merged (b,c,f,w) layout.
// Block = one 16-row q tile x full d=128 (8 waves, one 16-d tile each).
// P_lds [16][516]: rows 2064B (16B mult) -> B128 async; stride%64==4 ->
//   conflict-free A-frag reads.  V chunks of 32 k: Vt [128][36] rows 144B ->
//   B128 async; banks 4*(9L%16) distinct, cross-half 36*dL==2 impossible.
// Softmax: stage raw qk, 16x16 partial tree for rowmax/rowsum, exp in place,
// fold 1/rowsum into the epilogue (uniform per output row).
// ---------------------------------------------------------------------------
#define PPAD 516
#define VPAD 36
#define VCHUNK 32
__global__ __launch_bounds__(256) void av_softmax_wmma(
    const float* __restrict__ qk, const float* __restrict__ vp,
    float* __restrict__ am)
{
  __shared__ float P[16 * PPAD];
  __shared__ float Vt[128 * VPAD];
  __shared__ float red[16 * 16];
  __shared__ float rmax[16];
  __shared__ float rsum[16];
  const int tid  = threadIdx.x;
  const int wv   = tid >> 5, lane = tid & 31, L = lane & 15, hf = lane >> 4;
  const int qt   = blockIdx.x;                 // 0..31 (q tile)
  const int nh   = blockIdx.y;                 // 0..7
  const int bc   = blockIdx.z;                 // 0..15
  const int head = bc * NHn + nh;
  const float* qkb = qk + (size_t)head * Wn * Wn + (size_t)qt * 16 * Wn;
  const float* vb  = vp + (size_t)bc * Fn * Wn + (size_t)nh * HDn * Wn; // [d][kpos]
  float*       ab  = am + (size_t)bc * Fn * Wn + (size_t)nh * HDn * Wn; // [d][qpos]

  #pragma unroll
  for (int i = 0; i < 8; ++i) {                // 16x512 qk rows, B128 granules
    const int e = tid + i * 256;
    const int r = e >> 7, c4 = (e & 127) * 4;
    async_cp_b128(lds_off(&P[r * PPAD + c4]), &qkb[(size_t)r * Wn + c4]);
  }
  wait_async0();
  __syncthreads();
  {                                            // row-max partials (16x16)
    const int r = tid & 15, ch = tid >> 4;
    float mx = -3.402823466e38f;
    for (int i = 0; i < 32; ++i) mx = fmaxf(mx, P[r * PPAD + ch * 32 + i]);
    red[r * 16 + ch] = mx;
  }
  __syncthreads();
  if (tid < 16) {
    float mx = red[tid * 16];
    for (int i = 1; i < 16; ++i) mx = fmaxf(mx, red[tid * 16 + i]);
    rmax[tid] = mx;
  }
  __syncthreads();
  {                                            // exp in place + sum partials
    const int r = tid & 15, ch = tid >> 4;
    const float mx = rmax[r];
    float sm = 0.f;
    for (int i = 0; i < 32; ++i) {
      const float e = __expf(P[r * PPAD + ch * 32 + i] - mx);
      P[r * PPAD + ch * 32 + i] = e;
      sm += e;
    }
    red[r * 16 + ch] = sm;
  }
  __syncthreads();
  if (tid < 16) {
    float sm = 0.f;
    for (int i = 0; i < 16; ++i) sm += red[tid * 16 + i];
    rsum[tid] = 1.0f / sm;
  }

  v8f acc = vzero8();
  for (int kc = 0; kc < Wn; kc += VCHUNK) {
    __syncthreads();                           // WAR vs previous chunk
    #pragma unroll
    for (int i = 0; i < 4; ++i) {              // V chunk 128x32, B128 granules
      const int e = tid + i * 256;
      const int d = e >> 3, k4 = (e & 7) * 4;
      async_cp_b128(lds_off(&Vt[d * VPAD + k4]),
                    &vb[(size_t)d * Wn + kc + k4]);
    }
    wait_async0();
    __syncthreads();
    #pragma unroll
    for (int ks = 0; ks < VCHUNK / 4; ++ks) {
      const int k0 = ks * 4 + hf * 2;
      v2f af, bf;
      af.x = P[L * PPAD + kc + k0];          af.y = P[L * PPAD + kc + k0 + 1];
      bf.x = Vt[(wv * 16 + L) * VPAD + k0];  bf.y = Vt[(wv * 16 + L) * VPAD + k0 + 1];
      acc = wmma_f32(af, bf, acc);
    }
  }
  #pragma unroll
  for (int j = 0; j < 8; ++j) {                // D: M=q-in-tile, N=d-in-tile
    const int q = hf * 8 + j;
    const int d = wv * 16 + L;
    ab[(size_t)d * Wn + qt * 16 + q] = acc[j] * rsum[q];
  }
}

// ---------------------------------------------------------------------------
extern "C" void kernel_launch(void* const* d_in, const int* in_sizes, int n_in,
                              void* d_out, int out_size, void* d_ws, size_t ws_size,
                              hipStream_t stream) {
  const float* x    = (const float*)d_in[0];
  const float* prev = (const float*)d_in[1];
  const float* wqc  = (const float*)d_in[2];
  const float* bqc  = (const float*)d_in[3];
  const float* wql  = (const float*)d_in[4];
  const float* wkc  = (const float*)d_in[5];
  const float* bkc  = (const float*)d_in[6];
  const float* wkl  = (const float*)d_in[7];
  const float* wvc  = (const float*)d_in[8];
  const float* bvc  = (const float*)d_in[9];
  const float* wvl  = (const float*)d_in[10];
  const float* wol  = (const float*)d_in[11];

  const size_t XSZ = (size_t)Bn * Cn * Fn * Wn;   // 8,388,608 floats
  float* out   = (float*)d_out;                   // (B,C,F,W)
  float* qkout = out + XSZ;                       // (B,C,NH,W,W)

  float* ws = (float*)d_ws;                       // needs 6*XSZ*4 = 201 MB
  float* qc = ws + 0 * XSZ;
  float* kc = ws + 1 * XSZ;
  float* vc = ws + 2 * XSZ;
  float* qp = ws + 3 * XSZ;
  float* kp = ws + 4 * XSZ;
  float* vp = ws + 5 * XSZ;
  float* am = qc;                                 // qc dead after q projection

  conv3x3_qkv<<<dim3(2, 1024, 16), 256, 0, stream>>>(x, wqc, bqc, wkc, bkc,
                                                     wvc, bvc, qc, kc, vc);
  mc_linear_wmma<<<dim3(16, 4, 16), 256, 0, stream>>>(wql, qc, qp);
  mc_linear_wmma<<<dim3(16, 4, 16), 256, 0, stream>>>(wkl, kc, kp);
  mc_linear_wmma<<<dim3(16, 4, 16), 256, 0, stream>>>(wvl, vc, vp);
  rotary_inplace<<<32768, 256, 0, stream>>>(qp, kp);
  qk_wmma<<<dim3(8, 8, 128), 256, 0, stream>>>(qp, kp, prev, qkout);
  av_softmax_wmma<<<dim3(32, 8, 16), 256, 0, stream>>>(qkout, vp, am);
  mc_linear_wmma<<<dim3(16, 4, 16), 256, 0, stream>>>(wol, am, out);
}